// DynamicRoutingPooler_25606595019362
// MI455X (gfx1250) — compile-verified
//
#include <hip/hip_runtime.h>
#include <math.h>

#define B 8
#define S 2048
#define DIN 512
#define DOUT 512
#define C 8
#define CP 16               // capsule dim padded to one WMMA tile
#define ROUTING_ITERS 3
#define EPS 1e-8f
#define LN_EPS 1e-5f

typedef __attribute__((ext_vector_type(2))) float v2f;
typedef __attribute__((ext_vector_type(8))) float v8f;

// ---------------------------------------------------------------- init: rp = 1/S (cols<8) / 0 (pad), uT = 0
__global__ void k_init(float* __restrict__ rp, float* __restrict__ uT) {
    const int idx = blockIdx.x * blockDim.x + threadIdx.x;
    if (idx < B * S * CP) rp[idx] = ((idx & (CP - 1)) < C) ? (1.0f / (float)S) : 0.0f;
    if (idx < B * DIN * CP) uT[idx] = 0.0f;
}

// ---------------------------------------------------------------- M[b,c,i] = sum_s r[b,s,c] * F[b,s,i]
// One wave32 per 16x16 D tile: D[m=c(pad16), n=i], K = s in chunks of 4.
// A[m,k] = rp[s0+k, m]  (zero-padded columns -> no divergence)
// B[k,n] = F[s0+k, i0+n]
__global__ void k_wmma_M(const float* __restrict__ F, const float* __restrict__ rp,
                         float* __restrict__ M) {
    const int b    = blockIdx.x >> 5;          // 32 i-tiles per batch
    const int i0   = (blockIdx.x & 31) << 4;
    const int lane = threadIdx.x;
    const int half = lane >> 4;                // K split: k = g + 2*half
    const int lm   = lane & 15;                // m (A) / n (B)
    const float* Fb = F + (size_t)b * S * DIN;
    const float* rb = rp + (size_t)b * S * CP;

    v8f acc = {};
    for (int s0 = 0; s0 < S; s0 += 4) {
        const int k0 = s0 + 2 * half;
        v2f a, bb;
        a.x  = rb[(size_t)(k0 + 0) * CP + lm];
        a.y  = rb[(size_t)(k0 + 1) * CP + lm];
        bb.x = Fb[(size_t)(k0 + 0) * DIN + i0 + lm];
        bb.y = Fb[(size_t)(k0 + 1) * DIN + i0 + lm];
        acc = __builtin_amdgcn_wmma_f32_16x16x4_f32(false, a, false, bb,
                                                    (short)0, acc, false, false);
    }
    // D[m,n]: m = g + 8*half; only m(=c) < 8 is real
    if (half == 0) {
        float* Mb = M + (size_t)b * C * DIN;
#pragma unroll
        for (int g = 0; g < 8; ++g)
            Mb[(size_t)g * DIN + i0 + lm] = acc[g];
    }
}

// ---------------------------------------------------------------- agree[b,s,c] = sum_i F[b,s,i]*u[b,c,i] + bias[c]*vsum[b,c]
// One wave32 per 16x16 D tile: D[m=s, n=c(pad16)], K = i in chunks of 4.
// A[m,k] = F[s0+m, i0+k]   -> contiguous v2f per lane (global_load_b64)
// B[k,n] = uT[i0+k, n]     (zero-padded columns -> no divergence)
__global__ void k_wmma_agree(const float* __restrict__ F, const float* __restrict__ uT,
                             const float* __restrict__ bias, const float* __restrict__ vsum,
                             float* __restrict__ agree) {
    const int b    = blockIdx.x >> 7;          // 128 s-tiles per batch
    const int s0   = (blockIdx.x & 127) << 4;
    const int lane = threadIdx.x;
    const int half = lane >> 4;
    const int lm   = lane & 15;
    const float* Fb  = F + (size_t)b * S * DIN;
    const float* uTb = uT + (size_t)b * DIN * CP;
    const float* Frow = Fb + (size_t)(s0 + lm) * DIN;

    v8f acc = {};
    for (int i0 = 0; i0 < DIN; i0 += 4) {
        const int k0 = i0 + 2 * half;
        const v2f a = *(const v2f*)(Frow + k0);          // 8B-aligned: k0 even
        v2f bb;
        bb.x = uTb[(size_t)(k0 + 0) * CP + lm];
        bb.y = uTb[(size_t)(k0 + 1) * CP + lm];
        acc = __builtin_amdgcn_wmma_f32_16x16x4_f32(false, a, false, bb,
                                                    (short)0, acc, false, false);
    }
    if (lm < C) {
        const float bterm = bias[lm] * vsum[b * C + lm];
        float* ab = agree + (size_t)b * S * C;
#pragma unroll
        for (int g = 0; g < 8; ++g) {
            const int s = s0 + g + 8 * half;
            ab[(size_t)s * C + lm] = acc[g] + bterm;
        }
    }
}

// ---------------------------------------------------------------- block reduction (blockDim == 256)
__device__ __forceinline__ float block_reduce_sum(float val, float* sh) {
    const int tid = threadIdx.x;
    sh[tid] = val;
    __syncthreads();
    for (int off = 128; off > 0; off >>= 1) {
        if (tid < off) sh[tid] += sh[tid + off];
        __syncthreads();
    }
    const float res = sh[0];
    __syncthreads();
    return res;
}

// ---------------------------------------------------------------- s_in + squash -> v[b,c,d], vsum[b,c]
__global__ void k_sin_squash(const float* __restrict__ M, const float* __restrict__ W,
                             const float* __restrict__ bias,
                             float* __restrict__ v, float* __restrict__ vsum) {
    __shared__ float Msh[DIN];
    __shared__ float red[256];
    const int b   = blockIdx.x >> 3;
    const int c   = blockIdx.x & 7;
    const int tid = threadIdx.x;

    const float* Mb = M + ((size_t)b * C + c) * DIN;
    Msh[tid]       = Mb[tid];
    Msh[tid + 256] = Mb[tid + 256];
    __syncthreads();

    const int d0 = tid, d1 = tid + 256;
    float s0 = bias[c], s1 = bias[c];            // + bias * (sum_s r == 1)
    for (int i = 0; i < DIN; ++i) {
        const float m = Msh[i];
        const float* Wr = W + (size_t)i * DOUT * C + c;
        s0 += m * Wr[(size_t)d0 * C];
        s1 += m * Wr[(size_t)d1 * C];
    }
    const float ss = block_reduce_sum(s0 * s0 + s1 * s1, red);   // ||s_in||^2 over d
    const float n  = sqrtf(ss);
    const float scale = (ss / (ss + 1.0f)) / (n + EPS);
    const float v0 = scale * s0, v1 = scale * s1;

    float* vb = v + ((size_t)b * C + c) * DOUT;
    vb[d0] = v0;
    vb[d1] = v1;
    const float vs = block_reduce_sum(v0 + v1, red);
    if (tid == 0) vsum[b * C + c] = vs;
}

// ---------------------------------------------------------------- uT[b,i,c] = sum_d v[b,c,d] * W[i,d,c]   (pad cols stay 0)
__global__ void k_u(const float* __restrict__ v, const float* __restrict__ W,
                    float* __restrict__ uT) {
    __shared__ float vsh[DOUT];
    const int b   = blockIdx.x >> 3;
    const int c   = blockIdx.x & 7;
    const int tid = threadIdx.x;

    const float* vb = v + ((size_t)b * C + c) * DOUT;
    vsh[tid]       = vb[tid];
    vsh[tid + 256] = vb[tid + 256];
    __syncthreads();

    float* ub = uT + (size_t)b * DIN * CP;
#pragma unroll
    for (int ii = 0; ii < 2; ++ii) {
        const int i = tid + ii * 256;
        const float* Wr = W + (size_t)i * DOUT * C + c;
        float acc = 0.0f;
        for (int d = 0; d < DOUT; ++d) acc += vsh[d] * Wr[(size_t)d * C];
        ub[(size_t)i * CP + c] = acc;
    }
}

// ---------------------------------------------------------------- r = softmax_s(log(r+eps) + agree)  (on padded rp, cols<8 only)
__global__ void k_softmax(const float* __restrict__ agree, float* __restrict__ rp) {
    __shared__ float red[256];
    const int b   = blockIdx.x >> 3;
    const int c   = blockIdx.x & 7;
    const int tid = threadIdx.x;
    const float* ab = agree + (size_t)b * S * C;
    float* rb = rp + (size_t)b * S * CP;

    float logits[8];
    float lmax = -3.0e38f;
#pragma unroll
    for (int j = 0; j < 8; ++j) {
        const int s = tid + j * 256;
        const float l = logf(rb[(size_t)s * CP + c] + EPS) + ab[(size_t)s * C + c];
        logits[j] = l;
        lmax = fmaxf(lmax, l);
    }
    red[tid] = lmax;
    __syncthreads();
    for (int off = 128; off > 0; off >>= 1) {
        if (tid < off) red[tid] = fmaxf(red[tid], red[tid + off]);
        __syncthreads();
    }
    const float gmax = red[0];
    __syncthreads();

    float lsum = 0.0f;
#pragma unroll
    for (int j = 0; j < 8; ++j) {
        logits[j] = expf(logits[j] - gmax);
        lsum += logits[j];
    }
    const float inv = 1.0f / block_reduce_sum(lsum, red);
#pragma unroll
    for (int j = 0; j < 8; ++j) {
        const int s = tid + j * 256;
        rb[(size_t)s * CP + c] = logits[j] * inv;
    }
}

// ---------------------------------------------------------------- out[b,c,:] = LayerNorm_d(v[b,c,:])
__global__ void k_ln(const float* __restrict__ v, const float* __restrict__ gamma,
                     const float* __restrict__ beta, float* __restrict__ out) {
    __shared__ float red[256];
    const int b   = blockIdx.x >> 3;
    const int c   = blockIdx.x & 7;
    const int tid = threadIdx.x;
    const float* vb = v + ((size_t)b * C + c) * DOUT;

    const float x0 = vb[tid], x1 = vb[tid + 256];
    const float mean = block_reduce_sum(x0 + x1, red) * (1.0f / DOUT);
    const float d0 = x0 - mean, d1 = x1 - mean;
    const float var = block_reduce_sum(d0 * d0 + d1 * d1, red) * (1.0f / DOUT);
    const float inv = rsqrtf(var + LN_EPS);

    float* ob = out + ((size_t)b * C + c) * DOUT;
    ob[tid]       = d0 * inv * gamma[tid]       + beta[tid];
    ob[tid + 256] = d1 * inv * gamma[tid + 256] + beta[tid + 256];
}

// ----------------------------------------------------------------
extern "C" void kernel_launch(void* const* d_in, const int* in_sizes, int n_in,
                              void* d_out, int out_size, void* d_ws, size_t ws_size,
                              hipStream_t stream) {
    const float* F     = (const float*)d_in[0];  // [B,S,DIN]
    const float* W     = (const float*)d_in[1];  // [DIN,DOUT,C]
    const float* bias  = (const float*)d_in[2];  // [C]
    const float* gamma = (const float*)d_in[3];  // [DOUT]
    const float* beta  = (const float*)d_in[4];  // [DOUT]

    float* ws    = (float*)d_ws;
    float* rp    = ws;                  // B*S*CP     = 262144 (zero-padded capsule cols)
    float* agree = rp + B * S * CP;     // B*S*C      = 131072
    float* Mbuf  = agree + B * S * C;   // B*C*DIN    = 32768
    float* uT    = Mbuf + B * C * DIN;  // B*DIN*CP   = 65536 (zero-padded capsule cols)
    float* v     = uT + B * DIN * CP;   // B*C*DOUT   = 32768
    float* vsum  = v + B * C * DOUT;    // B*C        = 64

    k_init<<<(B * S * CP + 255) / 256, 256, 0, stream>>>(rp, uT);

    for (int it = 0; it < ROUTING_ITERS; ++it) {
        k_wmma_M<<<B * 32, 32, 0, stream>>>(F, rp, Mbuf);
        k_sin_squash<<<B * C, 256, 0, stream>>>(Mbuf, W, bias, v, vsum);
        if (it < ROUTING_ITERS - 1) {
            k_u<<<B * C, 256, 0, stream>>>(v, W, uT);
            k_wmma_agree<<<B * 128, 32, 0, stream>>>(F, uT, bias, vsum, agree);
            k_softmax<<<B * C, 256, 0, stream>>>(agree, rp);
        }
    }
    k_ln<<<B * C, 256, 0, stream>>>(v, gamma, beta, (float*)d_out);
}